// wvfn_9852654977036
// MI455X (gfx1250) — compile-verified
//
#include <hip/hip_runtime.h>
#include <stdint.h>

// ---------------------------------------------------------------------------
// Physics constants (match reference)
// ---------------------------------------------------------------------------
#define VB_F   0.1f
#define C00_F  0.28209479177387814f   // 0.5/sqrt(pi)
#define C10_F  0.4886025119029199f    // sqrt(3/(4pi))
#define C11_F  0.34549414947133544f   // sqrt(3/(8pi))

#define WALKERS_PER_BLK 128           // 8 waves * 16 walkers
#define TILE_DW (WALKERS_PER_BLK * 12)

typedef unsigned int u32x4 __attribute__((ext_vector_type(4)));
typedef int          i32x4 __attribute__((ext_vector_type(4)));
typedef int          i32x8 __attribute__((ext_vector_type(8)));
typedef float        v2f   __attribute__((ext_vector_type(2)));
typedef float        v8f   __attribute__((ext_vector_type(8)));

#ifndef __has_builtin
#define __has_builtin(x) 0
#endif

#if __has_builtin(__builtin_amdgcn_tensor_load_to_lds) && \
    __has_builtin(__builtin_amdgcn_s_wait_tensorcnt)
#define HAVE_TDM 1
#else
#define HAVE_TDM 0
#endif

#if __has_builtin(__builtin_amdgcn_wmma_f32_16x16x4_f32)
#define HAVE_WMMA 1
#else
#define HAVE_WMMA 0
#endif

// ---------------------------------------------------------------------------
// Prep kernel:
//  1) u[s][a] = sum_j C[n_s-1][l_s][m_s+l_s][a][j]   (complex, 5x4)
//  2) coef[16][40] into d_ws: rows 0..3 = {pr, pi, kr, ki} coefficients over
//     the 40-feature vector (e-major, 10 features per electron:
//     g=0:e1  1:e2  2:e2*x  3:e2*y  4:e2*z  5..9: same scaled by lap),
//     rows 4..15 zero (unused WMMA M rows).  -0.5 of K_psi folded into rows 2,3.
// ---------------------------------------------------------------------------
__global__ void prep_kernel(const float* __restrict__ C_re,
                            const float* __restrict__ C_im,
                            float* __restrict__ coef) {
  __shared__ float ur[5][4], ui[5][4];
  const int t = threadIdx.x;
  if (t < 20) {
    const int slice[5] = {0, 6, 9, 10, 11};   // ((n-1)*2+l)*3+(m+l)
    int s = t >> 2, a = t & 3;
    int base = slice[s] * 16 + a * 4;
    float sr = 0.f, si = 0.f;
#pragma unroll
    for (int j = 0; j < 4; ++j) { sr += C_re[base + j]; si += C_im[base + j]; }
    ur[s][a] = sr;
    ui[s][a] = si;
  }
  __syncthreads();

  for (int idx = t; idx < 16 * 40; idx += blockDim.x) {
    const int m = idx / 40;       // output row
    const int k = idx % 40;       // feature
    const int e = k / 10;         // electron
    const int f = k % 10;         // feature id
    float c = 0.f;
    if (m < 4) {
      const bool lapRow  = (m >= 2);
      const bool lapFeat = (f >= 5);
      const int  g = f % 5;
      if (lapRow == lapFeat) {
        float v;
        if ((m & 1) == 0) {       // pr-style row
          v = (g == 0) ? C00_F * ur[0][e]
            : (g == 1) ? C00_F * ur[1][e]
            : (g == 2) ? C11_F * (ur[2][e] - ur[4][e])
            : (g == 3) ? C11_F * (ui[2][e] + ui[4][e])
            :            C10_F * ur[3][e];
        } else {                  // pi-style row
          v = (g == 0) ? C00_F * ui[0][e]
            : (g == 1) ? C00_F * ui[1][e]
            : (g == 2) ? C11_F * (ui[2][e] - ui[4][e])
            : (g == 3) ? -C11_F * (ur[2][e] + ur[4][e])
            :            C10_F * ui[3][e];
        }
        c = lapRow ? -0.5f * v : v;
      }
    }
    coef[idx] = c;
  }
}

// ---------------------------------------------------------------------------
// Main kernel: 8 wave32 per block, each wave owns 16 walkers; both half-waves
// compute the same walkers' features so the WMMA B operand (lanes 0-15: K=0,1;
// lanes 16-31: K=2,3 of column N=lane&15) needs only value-cndmasks.
// Rs tile streamed into LDS by the Tensor Data Mover.
// ---------------------------------------------------------------------------
__global__ __launch_bounds__(256)
__attribute__((amdgpu_waves_per_eu(2)))
void walker_kernel(
    const float* __restrict__ Rs, const float* __restrict__ a_in,
    const float* __restrict__ coef, float* __restrict__ out, int N) {
  __shared__ float tile[TILE_DW];

  const int block0 = blockIdx.x * WALKERS_PER_BLK;
  int tile_elems = (N - block0) * 12;
  if (tile_elems > TILE_DW) tile_elems = TILE_DW;
  if (tile_elems < 0) tile_elems = 0;

#if HAVE_TDM
  if ((threadIdx.x >> 5) == 0 && tile_elems > 0) {
    // ---- Tensor DMA descriptor (ISA cdna5 §8.3-8.6), 1D tile of dwords ----
    uint64_t gaddr = (uint64_t)(uintptr_t)(Rs + (size_t)block0 * 12);
    uint32_t laddr = (uint32_t)(uintptr_t)(&tile[0]);   // addr[31:0] == LDS offset

    u32x4 g0;
    g0[0] = 1u;                                          // count=1, user mode
    g0[1] = laddr;                                       // lds_addr (bytes)
    g0[2] = (uint32_t)gaddr;                             // global_addr[31:0]
    g0[3] = (uint32_t)((gaddr >> 32) & 0x1FFFFFFull)     // global_addr[56:32]
          | (2u << 30);                                  // type=2 ("image")

    uint32_t te = (uint32_t)tile_elems;
    i32x8 g1;
    g1[0] = (int)(2u << 16);                 // mask=0, data_size=2 (4B)
    g1[1] = (int)((te & 0xFFFFu) << 16);     // tensor_dim0[15:0]
    g1[2] = (int)((te >> 16) | (1u << 16));  // tensor_dim0[31:16] | tensor_dim1=1
    g1[3] = (int)(te << 16);                 // tile_dim0
    g1[4] = 0;                               // tile_dim1=0 (1D), tile_dim2=0
    g1[5] = (int)te;                         // tensor_dim0_stride[31:0]
    g1[6] = 0;
    g1[7] = 0;

    i32x4 g2 = {0, 0, 0, 0};
    i32x4 g3 = {0, 0, 0, 0};

#if defined(__clang_major__) && (__clang_major__ >= 23)
    i32x8 g4 = {0, 0, 0, 0, 0, 0, 0, 0};
    __builtin_amdgcn_tensor_load_to_lds(g0, g1, g2, g3, g4, 0);
#else
    __builtin_amdgcn_tensor_load_to_lds(g0, g1, g2, g3, 0);
#endif
    __builtin_amdgcn_s_wait_tensorcnt(0);
  }
  asm volatile("" ::: "memory");
#else
  {
    const float4* __restrict__ src = (const float4*)(Rs + (size_t)block0 * 12);
    float4* dst = (float4*)tile;
    int n4 = tile_elems >> 2;
    for (int i = threadIdx.x; i < n4; i += 256) dst[i] = src[i];
  }
#endif
  __syncthreads();

  const int lane = (int)(threadIdx.x & 31);
  const int wave = (int)(threadIdx.x >> 5);
  const int wl   = lane & 15;                   // walker-in-wave (both halves)
  const int w    = block0 + wave * 16 + wl;

  // Wave-uniform exit for fully out-of-range waves (keeps EXEC all-ones for
  // surviving waves through the WMMAs, as the ISA requires).
  if (block0 + wave * 16 >= N) return;

  // ---- per-electron coordinates (lanes L and L+16 read the same walker) ----
  float X[4], Y[4], Z4[4];
  const float* p = &tile[(wave * 16 + wl) * 12];
#pragma unroll
  for (int e = 0; e < 4; ++e) {
    X[e]  = p[e * 3 + 0];
    Y[e]  = p[e * 3 + 1];
    Z4[e] = p[e * 3 + 2];
  }

  // ---- pair Coulomb term early (lets coords die before the WMMA region) ----
  float V = 0.f;
#pragma unroll
  for (int a = 0; a < 3; ++a) {
#pragma unroll
    for (int b = a + 1; b < 4; ++b) {
      float dx = X[a] - X[b], dy = Y[a] - Y[b], dz = Z4[a] - Z4[b];
      V -= VB_F / sqrtf(dx * dx + dy * dy + dz * dz);
    }
  }

  const float Zk = 1.0f / a_in[0];
  const float k0 = Zk;
  const float k1 = 0.5f * Zk;

  // ---- features, 10 per electron, in *distinct* small arrays so the
  //      lane<16 selects below stay value-selects (no gep-select -> no scratch)
  float fe1[4], fe2[4], fx[4], fy[4], fz[4];    // e1, e2, e2*x, e2*y, e2*z
  float g0f[4], g1f[4], gx[4], gy[4], gz[4];    // lap-scaled versions
#pragma unroll
  for (int e = 0; e < 4; ++e) {
    const float r2   = X[e] * X[e] + Y[e] * Y[e] + Z4[e] * Z4[e];
    const float r    = sqrtf(r2);
    const float ir   = 1.0f / r;
    const float e1   = expf(-k0 * r);
    const float e2   = expf(-k1 * r);
    const float lap0 = k0 * k0 - 2.0f * k0 * ir;
    const float lap1 = k1 * k1 - 2.0f * k1 * ir;
    const float lap2 = k1 * k1 - 4.0f * k1 * ir;
    fe1[e] = e1;
    fe2[e] = e2;
    fx[e]  = e2 * X[e];
    fy[e]  = e2 * Y[e];
    fz[e]  = e2 * Z4[e];
    g0f[e] = e1 * lap0;
    g1f[e] = e2 * lap1;
    gx[e]  = fx[e] * lap2;
    gy[e]  = fy[e] * lap2;
    gz[e]  = fz[e] * lap2;
  }

  float pr, pi_, kr, ki;
#if HAVE_WMMA
  // D(16x16) += A(16x4: coef chunk) x B(4x16: feature chunk), K-chunked by 4.
  // A: lanes 0-15 -> M=lane,    K=4c+{0,1}; lanes16-31 -> M=lane-16, K=4c+{2,3}
  // B: lanes 0-15 -> rows K=4c+{0,1}, N=lane; lanes16-31 -> rows K=4c+{2,3}
  const bool lo = (lane < 16);
  // B pairs per chunk: Bc0 = lo ? f[4c] : f[4c+2], Bc1 = lo ? f[4c+1] : f[4c+3]
  // flat order per electron: e1, e2, e2x, e2y, e2z, l0, l1, l2x, l2y, l2z
  float B00 = lo ? fe1[0] : fx[0];   float B01 = lo ? fe2[0] : fy[0];
  float B10 = lo ? fz[0]  : g1f[0];  float B11 = lo ? g0f[0] : gx[0];
  float B20 = lo ? gy[0]  : fe1[1];  float B21 = lo ? gz[0]  : fe2[1];
  float B30 = lo ? fx[1]  : fz[1];   float B31 = lo ? fy[1]  : g0f[1];
  float B40 = lo ? g1f[1] : gy[1];   float B41 = lo ? gx[1]  : gz[1];
  float B50 = lo ? fe1[2] : fx[2];   float B51 = lo ? fe2[2] : fy[2];
  float B60 = lo ? fz[2]  : g1f[2];  float B61 = lo ? g0f[2] : gx[2];
  float B70 = lo ? gy[2]  : fe1[3];  float B71 = lo ? gz[2]  : fe2[3];
  float B80 = lo ? fx[3]  : fz[3];   float B81 = lo ? fy[3]  : g0f[3];
  float B90 = lo ? g1f[3] : gy[3];   float B91 = lo ? gx[3]  : gz[3];

  const int   ksel = lo ? 0 : 2;
  const float* ca  = coef + wl * 40 + ksel;

  v8f D = {0.f, 0.f, 0.f, 0.f, 0.f, 0.f, 0.f, 0.f};
#define WMMA_STEP(c, b0, b1)                                                  \
  {                                                                           \
    v2f A; A[0] = ca[(c) * 4 + 0]; A[1] = ca[(c) * 4 + 1];                    \
    v2f B; B[0] = (b0); B[1] = (b1);                                          \
    D = __builtin_amdgcn_wmma_f32_16x16x4_f32(false, A, false, B,             \
                                              (short)0, D, false, false);     \
  }
  WMMA_STEP(0, B00, B01)
  WMMA_STEP(1, B10, B11)
  WMMA_STEP(2, B20, B21)
  WMMA_STEP(3, B30, B31)
  WMMA_STEP(4, B40, B41)
  WMMA_STEP(5, B50, B51)
  WMMA_STEP(6, B60, B61)
  WMMA_STEP(7, B70, B71)
  WMMA_STEP(8, B80, B81)
  WMMA_STEP(9, B90, B91)
#undef WMMA_STEP
  // Rows 0..3 at lanes 0-15 hold {pr, pi, kr, ki} for walker N=lane.
  pr  = D[0];
  pi_ = D[1];
  kr  = D[2];
  ki  = D[3];
#else
  {
    float feat[40];
#pragma unroll
    for (int e = 0; e < 4; ++e) {
      feat[e * 10 + 0] = fe1[e];  feat[e * 10 + 1] = fe2[e];
      feat[e * 10 + 2] = fx[e];   feat[e * 10 + 3] = fy[e];
      feat[e * 10 + 4] = fz[e];   feat[e * 10 + 5] = g0f[e];
      feat[e * 10 + 6] = g1f[e];  feat[e * 10 + 7] = gx[e];
      feat[e * 10 + 8] = gy[e];   feat[e * 10 + 9] = gz[e];
    }
    float a0 = 0.f, a1 = 0.f, a2 = 0.f, a3 = 0.f;
#pragma unroll
    for (int k = 0; k < 40; ++k) {
      const float f = feat[k];
      a0 += coef[0 * 40 + k] * f;
      a1 += coef[1 * 40 + k] * f;
      a2 += coef[2 * 40 + k] * f;
      a3 += coef[3 * 40 + k] * f;
    }
    pr = a0; pi_ = a1; kr = a2; ki = a3;
  }
#endif

  // ---- H_psi = K_psi + V*psi ; out1 = conj(psi)*H_psi/VB^2 ; out2 = |psi|^2
  if (lane < 16 && w < N) {
    const float Hr = kr + V * pr;
    const float Hi = ki + V * pi_;
    const float inv_vb2 = 1.0f / (VB_F * VB_F);   // 100
    out[2 * w]     = (pr * Hr + pi_ * Hi) * inv_vb2;
    out[2 * w + 1] = (pr * Hi - pi_ * Hr) * inv_vb2;
    out[2 * N + w] = pr * pr + pi_ * pi_;
  }
}

// ---------------------------------------------------------------------------
// Launch
// ---------------------------------------------------------------------------
extern "C" void kernel_launch(void* const* d_in, const int* in_sizes, int n_in,
                              void* d_out, int out_size, void* d_ws, size_t ws_size,
                              hipStream_t stream) {
  (void)n_in; (void)out_size; (void)ws_size;
  const float* Rs   = (const float*)d_in[0];
  const float* a_in = (const float*)d_in[1];
  const float* C_re = (const float*)d_in[2];
  const float* C_im = (const float*)d_in[3];
  float* coef = (float*)d_ws;        // 640 floats
  float* out  = (float*)d_out;

  const int N = in_sizes[0] / 12;    // walkers

  prep_kernel<<<1, 256, 0, stream>>>(C_re, C_im, coef);

  const int blocks = (N + WALKERS_PER_BLK - 1) / WALKERS_PER_BLK;
  walker_kernel<<<blocks, 256, 0, stream>>>(Rs, a_in, coef, out, N);
}